// BUIR_NB_54640573939822
// MI455X (gfx1250) — compile-verified
//
#include <hip/hip_runtime.h>
#include <hip/hip_bf16.h>

typedef __attribute__((ext_vector_type(2))) float v2f;
typedef __attribute__((ext_vector_type(8))) float v8f;

#define DEMB 64

// -------------------- init: acc = xcur = concat(user_emb, item_emb) --------------------
__global__ __launch_bounds__(256) void k_init_ego(const float* __restrict__ ue,
                                                  const float* __restrict__ ie,
                                                  float* __restrict__ acc,
                                                  float* __restrict__ xcur,
                                                  int usz, int total) {
    int i = blockIdx.x * 256 + threadIdx.x;
    if (i >= total) return;
    float v = (i < usz) ? ue[i] : ie[i - usz];
    acc[i]  = v;
    xcur[i] = v;
}

// -------------------- zero a buffer (float4-wide) --------------------
__global__ __launch_bounds__(256) void k_zero_f4(float4* __restrict__ p, int n4) {
    int i = blockIdx.x * 256 + threadIdx.x;
    if (i >= n4) return;
    float4 z; z.x = 0.f; z.y = 0.f; z.z = 0.f; z.w = 0.f;
    p[i] = z;
}

// -------------------- acc += x (float4-wide) --------------------
__global__ __launch_bounds__(256) void k_axpy_f4(float4* __restrict__ acc,
                                                 const float4* __restrict__ x, int n4) {
    int i = blockIdx.x * 256 + threadIdx.x;
    if (i >= n4) return;
    float4 a = acc[i];
    float4 b = x[i];
    a.x += b.x; a.y += b.y; a.z += b.z; a.w += b.w;
    acc[i] = a;
}

// -------------------- SpMM scatter: one wave32 per nonzero --------------------
// xnext[row,:] += val * xcur[col,:]   (64 floats -> 2 per lane, f32 atomics into L2)
__global__ __launch_bounds__(256) void k_spmm_scatter(const int* __restrict__ row,
                                                      const int* __restrict__ col,
                                                      const float* __restrict__ val,
                                                      const float* __restrict__ xcur,
                                                      float* __restrict__ xnext,
                                                      int nnz) {
    long long gid = (long long)blockIdx.x * 256 + threadIdx.x;
    int e    = (int)(gid >> 5);
    int lane = (int)(gid & 31);
    if (e >= nnz) return;
    int   r = row[e];           // wave-uniform address -> single L2 request
    int   c = col[e];
    float v = val[e];
    const float2* src = (const float2*)(xcur + (size_t)c * DEMB);
    float2 x = src[lane];
    float* dst = xnext + (size_t)r * DEMB + lane * 2;
    atomicAdd(dst,     v * x.x);   // global_atomic_add_f32, L2-resident target
    atomicAdd(dst + 1, v * x.y);
}

// -------------------- gather + mean(/4) for target branch --------------------
__global__ __launch_bounds__(256) void k_gather_scale(const float* __restrict__ acc,
                                                      const int* __restrict__ idx,
                                                      float* __restrict__ out,
                                                      int base, int total) {
    int i = blockIdx.x * 256 + threadIdx.x;
    if (i >= total) return;
    int b = i >> 6;
    int d = i & (DEMB - 1);
    out[i] = acc[(size_t)(idx[b] + base) * DEMB + d] * 0.25f;
}

// -------------------- prediction head: fused gather + (x/4) @ W^T + b --------------------
// One wave per 16x16 output tile; K=64 done as 16 chained V_WMMA_F32_16X16X4_F32.
// fp32 A layout (16x4): lanes 0-15 hold {K, K+1}, lanes 16-31 hold {K+2, K+3} for M=lane&15.
// fp32 B layout (4x16): VGPR0 = row K (+2 for hi lanes), VGPR1 = row K+1 (+2), N=lane&15.
// C/D layout: VGPR j -> M = j + 8*(lane>>4), N = lane&15.
__global__ __launch_bounds__(128) void k_pred_gemm_wmma(const float* __restrict__ acc,
                                                        const int* __restrict__ idx,
                                                        const float* __restrict__ w,
                                                        const float* __restrict__ bias,
                                                        float* __restrict__ out,
                                                        int base, int mtiles) {
    int waveInBlock = threadIdx.x >> 5;
    int lane        = threadIdx.x & 31;
    int tile = blockIdx.x * 4 + waveInBlock;   // 4 n-tiles per m-row (D=64)
    int mt = tile >> 2;
    int nt = tile & 3;
    if (mt >= mtiles) return;                  // whole wave exits together (EXEC stays all-1s)

    int mrow = lane & 15;
    int hi   = lane >> 4;                      // 0: K,K+1   1: K+2,K+3
    int m_global = mt * 16 + mrow;
    int r = idx[m_global] + base;              // batch gather (+U offset for items)
    const float* arow = acc + (size_t)r * DEMB;
    int n = nt * 16 + mrow;
    const float* wrow = w + (size_t)n * DEMB;  // B[k][n] = pred_w[n][k] (u @ W^T)

    v8f c = {0.f, 0.f, 0.f, 0.f, 0.f, 0.f, 0.f, 0.f};
#pragma unroll
    for (int k0 = 0; k0 < DEMB; k0 += 4) {
        const float* ap = arow + k0 + 2 * hi;
        const float* bp = wrow + k0 + 2 * hi;
        v2f a, b;
        a[0] = ap[0] * 0.25f;  a[1] = ap[1] * 0.25f;   // fold the /(L+1) mean into A
        b[0] = bp[0];          b[1] = bp[1];
        c = __builtin_amdgcn_wmma_f32_16x16x4_f32(false, a, false, b,
                                                  (short)0, c, false, false);
    }

    float bv = bias[n];
#pragma unroll
    for (int j = 0; j < 8; ++j) {
        int orow = mt * 16 + j + 8 * hi;
        out[(size_t)orow * DEMB + n] = c[j] + bv;
    }
}

// -------------------- host side --------------------
extern "C" void kernel_launch(void* const* d_in, const int* in_sizes, int n_in,
                              void* d_out, int out_size, void* d_ws, size_t ws_size,
                              hipStream_t stream) {
    const float* ue_on = (const float*)d_in[0];
    const float* ie_on = (const float*)d_in[1];
    const float* ue_tg = (const float*)d_in[2];
    const float* ie_tg = (const float*)d_in[3];
    const float* aval  = (const float*)d_in[4];
    const float* predw = (const float*)d_in[5];
    const float* predb = (const float*)d_in[6];
    const int*   arow  = (const int*)d_in[7];
    const int*   acol  = (const int*)d_in[8];
    const int*   uidx  = (const int*)d_in[9];
    const int*   iidx  = (const int*)d_in[10];

    const int D   = DEMB;
    const int U   = in_sizes[0] / D;
    const int I   = in_sizes[1] / D;
    const int N   = U + I;
    const int NNZ = in_sizes[4];
    const int B   = in_sizes[9];
    const int L   = 3;

    const size_t nd = (size_t)N * D;
    float* acc_on = (float*)d_ws;          // [N,64]
    float* acc_tg = acc_on + nd;           // [N,64]
    float* xa     = acc_tg + nd;           // ping
    float* xb     = xa + nd;               // pong

    const int total   = (int)nd;
    const int n4      = total / 4;
    const unsigned gInit = (unsigned)((total + 255) / 256);
    const unsigned gF4   = (unsigned)((n4 + 255) / 256);
    const unsigned gSpmm = (unsigned)(((long long)NNZ * 32 + 255) / 256);

    // ---- two LightGCN encodes (online, target), ping-pong x buffers ----
    const float* ue[2] = {ue_on, ue_tg};
    const float* ie[2] = {ie_on, ie_tg};
    float* accs[2]     = {acc_on, acc_tg};
    for (int e = 0; e < 2; ++e) {
        float* xcur = xa;
        float* xnext = xb;
        k_init_ego<<<gInit, 256, 0, stream>>>(ue[e], ie[e], accs[e], xcur, U * D, total);
        for (int l = 0; l < L; ++l) {
            k_zero_f4<<<gF4, 256, 0, stream>>>((float4*)xnext, n4);
            k_spmm_scatter<<<gSpmm, 256, 0, stream>>>(arow, acol, aval, xcur, xnext, NNZ);
            k_axpy_f4<<<gF4, 256, 0, stream>>>((float4*)accs[e], (const float4*)xnext, n4);
            float* t = xcur; xcur = xnext; xnext = t;
        }
    }

    // ---- outputs: (u_pred, u_target, i_pred, i_target), each [B,64] ----
    float* out    = (float*)d_out;
    float* u_pred = out;
    float* u_tgt  = out + (size_t)B * D;
    float* i_pred = u_tgt + (size_t)B * D;
    float* i_tgt  = i_pred + (size_t)B * D;

    const int mtiles = B / 16;                 // 256
    const unsigned gGemm = (unsigned)mtiles;   // 4 waves/block * 4 n-tiles = one m-row/block
    k_pred_gemm_wmma<<<gGemm, 128, 0, stream>>>(acc_on, uidx, predw, predb, u_pred, 0, mtiles);
    k_pred_gemm_wmma<<<gGemm, 128, 0, stream>>>(acc_on, iidx, predw, predb, i_pred, U, mtiles);

    const int bt = B * D;
    const unsigned gGather = (unsigned)((bt + 255) / 256);
    k_gather_scale<<<gGather, 256, 0, stream>>>(acc_tg, uidx, u_tgt, 0, bt);
    k_gather_scale<<<gGather, 256, 0, stream>>>(acc_tg, iidx, i_tgt, U, bt);
}